// RiemannianGeometry_18116172054532
// MI455X (gfx1250) — compile-verified
//
#include <hip/hip_runtime.h>

#define D 8
#define NP_P 8
#define HID 64
#define OUTW 512          // D^3
#define EPS_FD 1e-4f
#define EPS_PD 1e-6f
#define YSTRIDE 520       // 512 + pad to dodge LDS bank conflicts

typedef __attribute__((ext_vector_type(2))) float v2f;
typedef __attribute__((ext_vector_type(8))) float v8f;

// ---------------------------------------------------------------------------
// Kernel 1: layer-1 + central difference, folded.
// thread = (point, h):  base = x . W1[:,h] + b1[h]
//   d[(point*8+k)*64+h] = (tanh(base+eps*W1[k,h]) - tanh(base-eps*W1[k,h]))/(2 eps)
// ---------------------------------------------------------------------------
__global__ __launch_bounds__(256)
void k_layer1_diff(const float* __restrict__ pos,
                   const float* __restrict__ W1,
                   const float* __restrict__ b1,
                   float* __restrict__ Dall, int npoints)
{
    int tid = blockIdx.x * blockDim.x + threadIdx.x;
    int point = tid / HID;
    int h = tid % HID;
    if (point >= npoints) return;

    const float* x = pos + point * D;
    float base = b1[h];
#pragma unroll
    for (int i = 0; i < D; ++i) base += x[i] * W1[i * HID + h];

    const float inv2e = 0.5f / EPS_FD;
#pragma unroll
    for (int k = 0; k < D; ++k) {
        float w = W1[k * HID + h] * EPS_FD;
        float d = (tanhf(base + w) - tanhf(base - w)) * inv2e;
        Dall[(point * D + k) * HID + h] = d;
    }
}

// ---------------------------------------------------------------------------
// Kernel 2: Y = Dall(65536x64) @ W2(64x512) via V_WMMA_F32_16X16X4_F32,
// then per-row Riemann contraction + per-point metric denominator.
// One workgroup (4 waves, 128 thr) = 16 rows = 2 points.
// Wave w computes N-tiles [8w, 8w+8); 16 K-steps of K=4 each.
// ---------------------------------------------------------------------------
__global__ __launch_bounds__(128)
void k_gemm_contract(const float* __restrict__ Dall,
                     const float* __restrict__ W2,
                     const float* __restrict__ u,
                     const float* __restrict__ v,
                     const float* __restrict__ mc,   // metric_chol (P,D,D)
                     float* __restrict__ out)
{
    __shared__ float sY[16 * YSTRIDE];   // Y tile, 16 rows x 512
    __shared__ float sM[16][HID];        // m[row][a*8+b]
    __shared__ float sC[16];             // per-row contribution

    const int wg   = blockIdx.x;          // row-group id
    const int lane = threadIdx.x & 31;
    const int wave = threadIdx.x >> 5;
    const int R0   = wg * 16;

    // ---- A fragments: 16 K-steps, kept in registers, reused by 8 N-tiles ----
    // A(16x4) f32 layout: M = lane%16; VGPR0/1 = K {0,1} (lanes 0-15) / {2,3} (16-31)
    const int mrow  = R0 + (lane & 15);
    const int kbase = (lane < 16) ? 0 : 2;
    v2f afrag[16];
#pragma unroll
    for (int kk = 0; kk < 16; ++kk) {
        const float* ap = Dall + mrow * HID + kk * 4 + kbase;
        afrag[kk].x = ap[0];
        afrag[kk].y = ap[1];
    }

    // ---- GEMM ----
    for (int t = 0; t < 8; ++t) {
        const int ntile = wave * 8 + t;
        const int col   = ntile * 16 + (lane & 15);
        v8f acc = {};
#pragma unroll
        for (int kk = 0; kk < 16; ++kk) {
            // B(4x16) f32: N = lane%16; VGPR0/1 = K {0,1} / {2,3} mirroring A
            v2f bfrag;
            const float* bp = W2 + (kk * 4 + kbase) * OUTW + col;
            bfrag.x = bp[0];
            bfrag.y = bp[OUTW];
            acc = __builtin_amdgcn_wmma_f32_16x16x4_f32(
                false, afrag[kk], false, bfrag, (short)0, acc, false, false);
        }
        // C/D layout: VGPR j -> M = j (lanes 0-15) / j+8 (lanes 16-31); N = lane%16
        const int rbase = (lane < 16) ? 0 : 8;
#pragma unroll
        for (int j = 0; j < 8; ++j)
            sY[(rbase + j) * YSTRIDE + col] = acc[j];
    }
    __syncthreads();

    // ---- epilogue 1: thread = (row, a); m[a,b] = sum_i y[i,a,b] * u_i ----
    {
        const int row   = threadIdx.x >> 3;
        const int a     = threadIdx.x & 7;
        const int point = wg * 2 + (row >> 3);
        const float* up = u + point * D;
#pragma unroll
        for (int b = 0; b < 8; ++b) {
            float s = 0.f;
#pragma unroll
            for (int i = 0; i < 8; ++i)
                s += sY[row * YSTRIDE + i * 64 + a * 8 + b] * up[i];
            sM[row][a * 8 + b] = s;
        }
    }
    __syncthreads();

    // ---- epilogue 2: one thread per row -> masked Riemann scalar ----
    if (threadIdx.x < 16) {
        const int row   = threadIdx.x;
        const int point = wg * 2 + (row >> 3);
        const int k     = row & 7;
        const float* up = u + point * D;
        const float* vp = v + point * D;
        float P1[8], P2[8];
#pragma unroll
        for (int b = 0; b < 8; ++b) {
            float s = 0.f;
#pragma unroll
            for (int a = 0; a < 8; ++a) s += sM[row][a * 8 + b] * up[a];
            P1[b] = s;
        }
#pragma unroll
        for (int a = 0; a < 8; ++a) {
            float s = 0.f;
#pragma unroll
            for (int b = 0; b < 8; ++b) s += sM[row][a * 8 + b] * up[b];
            P2[a] = s;
        }
        float S1 = 0.f, S2 = 0.f;
#pragma unroll
        for (int i = 0; i < 8; ++i) { S1 += P1[i] * vp[i]; S2 += P2[i] * vp[i]; }
        const float term1 = S1 - P1[k] * vp[k];     // mask l != k
        const float term2 = S2 - P2[k] * vp[k];     // symmetrized partner
        sC[row] = vp[k] * 0.5f * (term1 + term2);
    }
    __syncthreads();

    // ---- epilogue 3: one thread per point -> denominator + final divide ----
    if (threadIdx.x < 2) {
        const int point = wg * 2 + (int)threadIdx.x;
        float num = 0.f;
#pragma unroll
        for (int r = 0; r < 8; ++r) num += sC[threadIdx.x * 8 + r];

        const int p = point & (NP_P - 1);           // point % P
        const float* L = mc + p * 64;               // tril
        float g[8][8];
#pragma unroll
        for (int i = 0; i < 8; ++i)
            for (int j = 0; j < 8; ++j) {
                float s = 0.f;
                const int mn = i < j ? i : j;
                for (int m2 = 0; m2 <= mn; ++m2) s += L[i * 8 + m2] * L[j * 8 + m2];
                g[i][j] = s + ((i == j) ? EPS_PD : 0.f);
            }
        const float* up = u + point * D;
        const float* vp = v + point * D;
        float guu = 0.f, gvv = 0.f, guv = 0.f;
#pragma unroll
        for (int i = 0; i < 8; ++i)
            for (int j = 0; j < 8; ++j) {
                guu += up[i] * g[i][j] * up[j];
                gvv += vp[i] * g[i][j] * vp[j];
                guv += up[i] * g[i][j] * vp[j];
            }
        float den = guu * gvv - guv * guv;
        if (den < 1e-8f) den = 1e-8f;
        out[point] = num / den;
    }
}

// ---------------------------------------------------------------------------
extern "C" void kernel_launch(void* const* d_in, const int* in_sizes, int n_in,
                              void* d_out, int out_size, void* d_ws, size_t ws_size,
                              hipStream_t stream)
{
    const float* pos = (const float*)d_in[0];   // (B,T,P,D)
    const float* u   = (const float*)d_in[1];
    const float* v   = (const float*)d_in[2];
    const float* mc  = (const float*)d_in[3];   // (P,D,D)
    const float* W1  = (const float*)d_in[4];   // (D,HID)
    const float* b1  = (const float*)d_in[5];
    const float* W2  = (const float*)d_in[6];   // (HID, D^3)
    // d_in[7] = b2: cancels in the central difference — unused.
    float* out = (float*)d_out;

    const int npoints = in_sizes[0] / D;        // B*T*P = 8192
    float* Dall = (float*)d_ws;                 // (npoints*8) x 64 fp32 = 16 MB

    const int t1 = npoints * HID;
    k_layer1_diff<<<(t1 + 255) / 256, 256, 0, stream>>>(pos, W1, b1, Dall, npoints);

    const int ngroups = (npoints * D) / 16;     // 16 rows per WG
    k_gemm_contract<<<ngroups, 128, 0, stream>>>(Dall, W2, u, v, mc, out);
}